// SimpleConvGCN_53171695124875
// MI455X (gfx1250) — compile-verified
//
#include <hip/hip_runtime.h>
#include <hip/hip_bf16.h>
#include <math.h>

// GCNConv: out = scatter_add(norm * (x@W)[row] -> col) + selfloop + b
#define IN_DIM  512
#define OUT_DIM 16

typedef __attribute__((ext_vector_type(2))) float v2f;
typedef __attribute__((ext_vector_type(8))) float v8f;

// ---------------------------------------------------------------------------
// init: zero out[], deg[] = 1 (self loop included in degree)
// ---------------------------------------------------------------------------
__global__ __launch_bounds__(256)
void gcn_init(float* __restrict__ out, unsigned* __restrict__ deg, int n) {
    int i = blockIdx.x * 256 + threadIdx.x;
    int total = n * OUT_DIM;
    if (i < total) out[i] = 0.0f;
    if (i < n)     deg[i] = 1u;
}

// ---------------------------------------------------------------------------
// degree: deg[col[e]] += 1
// ---------------------------------------------------------------------------
__global__ __launch_bounds__(256)
void gcn_degree(const int* __restrict__ col, unsigned* __restrict__ deg, int n_edges) {
    int e = blockIdx.x * 256 + threadIdx.x;
    if (e < n_edges) atomicAdd(&deg[col[e]], 1u);
}

// ---------------------------------------------------------------------------
// dinv[i] = rsqrt(deg[i])  (deg >= 1 always due to self loop)
// ---------------------------------------------------------------------------
__global__ __launch_bounds__(256)
void gcn_dinv(const unsigned* __restrict__ deg, float* __restrict__ dinv, int n) {
    int i = blockIdx.x * 256 + threadIdx.x;
    if (i < n) dinv[i] = rsqrtf((float)deg[i]);
}

// ---------------------------------------------------------------------------
// h = x @ W via V_WMMA_F32_16X16X4_F32 (full f32 precision, memory-bound).
// Block = 256 threads = 8 waves; block covers 128 rows, each wave a 16-row
// M tile x the single 16-col N tile. K staged through a DOUBLE-BUFFERED LDS
// tile filled with async global->LDS copies (ASYNCcnt), overlapping the next
// chunk's HBM traffic with the current chunk's WMMAs.
// Padded row stride 36 floats = 144B: 16B aligned, conflict-free column reads.
// ---------------------------------------------------------------------------
#define GEMM_ROWS 128
#define KC 32
#define KP 36
#define NCHUNK (IN_DIM / KC)   // 16

__device__ __forceinline__ void async_b128(unsigned lds_off, unsigned long long gaddr) {
    asm volatile("global_load_async_to_lds_b128 %0, %1, off"
                 :: "v"(lds_off), "v"(gaddr) : "memory");
}

__global__ __launch_bounds__(256)
void gcn_gemm_wmma(const float* __restrict__ x, const float* __restrict__ W,
                   float* __restrict__ h, int n) {
    __shared__ float tile[2][GEMM_ROWS * KP];

    const int tid  = threadIdx.x;
    const int wave = tid >> 5;      // 0..7
    const int lane = tid & 31;
    const int hi   = lane >> 4;     // 0: lanes 0-15, 1: lanes 16-31
    const int lm   = lane & 15;
    const int brow = blockIdx.x * GEMM_ROWS;

    // This thread's 4 staging slots (16B each): 128 rows x 8 float4 per chunk.
    unsigned long long gsrc[4];
    unsigned ldsoff[2][4];
    #pragma unroll
    for (int s = 0; s < 4; ++s) {
        int slot = tid + s * 256;
        int r    = slot >> 3;           // 8 float4 per row
        int cq   = (slot & 7) << 2;     // float column within chunk
        int gr   = brow + r;
        if (gr > n - 1) gr = n - 1;     // clamp: OOB rows produce garbage that is never stored
        gsrc[s]      = (unsigned long long)(x + (size_t)gr * IN_DIM + cq);
        ldsoff[0][s] = (unsigned)(uintptr_t)&tile[0][r * KP + cq];
        ldsoff[1][s] = (unsigned)(uintptr_t)&tile[1][r * KP + cq];
    }

    v8f acc = {0.f, 0.f, 0.f, 0.f, 0.f, 0.f, 0.f, 0.f};

    // Prologue: async-prefetch chunk 0 into buffer 0
    #pragma unroll
    for (int s = 0; s < 4; ++s) async_b128(ldsoff[0][s], gsrc[s]);

    for (int c = 0; c < NCHUNK; ++c) {
        const int cur = c & 1;
        // my chunk-c copies complete, then barrier => everyone's data is in LDS
        asm volatile("s_wait_asynccnt 0x0" ::: "memory");
        __syncthreads();
        // prefetch chunk c+1 into the other buffer (overlaps the WMMAs below)
        if (c + 1 < NCHUNK) {
            #pragma unroll
            for (int s = 0; s < 4; ++s)
                async_b128(ldsoff[cur ^ 1][s], gsrc[s] + (unsigned long long)(c + 1) * KC * sizeof(float));
        }

        const float* at  = &tile[cur][(wave * 16 + lm) * KP + 2 * hi];
        const float* wp0 = W + (size_t)(c * KC + 2 * hi) * OUT_DIM + lm;
        #pragma unroll
        for (int kk = 0; kk < KC; kk += 4) {
            // A fragment (16x4): lane half 'hi' holds K = kk+2*hi, kk+2*hi+1
            v2f a = *(const v2f*)(at + kk);
            // B fragment (4x16): same K split; lane lm = output column
            v2f b;
            b.x = wp0[kk * OUT_DIM];
            b.y = wp0[kk * OUT_DIM + OUT_DIM];
            acc = __builtin_amdgcn_wmma_f32_16x16x4_f32(
                false, a, false, b, (short)0, acc, false, false);
        }
    }

    // D layout: VGPR r -> row r (lanes 0-15) / row 8+r (lanes 16-31), col = lm
    const int wrow = brow + wave * 16;
    #pragma unroll
    for (int r = 0; r < 8; ++r) {
        int grow = wrow + r + 8 * hi;
        if (grow < n) h[(size_t)grow * OUT_DIM + lm] = acc[r];
    }
}

// ---------------------------------------------------------------------------
// scatter: 16 lanes per edge -> coalesced gather of h[row][:] (L2-resident,
// 6.4MB) and coalesced global_atomic_add_f32 into out[col][:] (also L2)
// ---------------------------------------------------------------------------
__global__ __launch_bounds__(256)
void gcn_scatter(const int* __restrict__ ei, const float* __restrict__ h,
                 const float* __restrict__ dinv, float* __restrict__ out,
                 int n_edges) {
    int t = blockIdx.x * 256 + threadIdx.x;
    int e = t >> 4;
    int j = t & 15;
    if (e >= n_edges) return;
    int r = ei[e];               // source
    int c = ei[n_edges + e];     // target
    float w = dinv[r] * dinv[c];
    unsafeAtomicAdd(&out[c * OUT_DIM + j], w * h[r * OUT_DIM + j]);
}

// ---------------------------------------------------------------------------
// finalize: self-loop term (norm = dinv[i]^2) + bias
// ---------------------------------------------------------------------------
__global__ __launch_bounds__(256)
void gcn_finalize(const float* __restrict__ h, const float* __restrict__ dinv,
                  const float* __restrict__ b, float* __restrict__ out, int n) {
    int i = blockIdx.x * 256 + threadIdx.x;
    if (i >= n * OUT_DIM) return;
    int node = i >> 4;
    int j = i & 15;
    float d = dinv[node];
    out[i] += d * d * h[i] + b[j];
}

// ---------------------------------------------------------------------------
extern "C" void kernel_launch(void* const* d_in, const int* in_sizes, int n_in,
                              void* d_out, int out_size, void* d_ws, size_t ws_size,
                              hipStream_t stream) {
    const float* x  = (const float*)d_in[0];
    const int*   ei = (const int*)d_in[1];      // [2, E] row-major (int32)
    const float* W  = (const float*)d_in[2];    // [512, 16] row-major
    const float* b  = (const float*)d_in[3];    // [16]

    const int n = in_sizes[0] / IN_DIM;         // 100000
    const int E = in_sizes[1] / 2;              // 3200000

    float* out = (float*)d_out;

    // workspace layout: h [n*16 f32] | deg [n u32] | dinv [n f32]  (~7.2 MB)
    float*    h    = (float*)d_ws;
    unsigned* deg  = (unsigned*)((char*)d_ws + (size_t)n * OUT_DIM * sizeof(float));
    float*    dinv = (float*)((char*)deg + (size_t)n * sizeof(unsigned));

    const int total_out = n * OUT_DIM;

    gcn_init<<<(total_out + 255) / 256, 256, 0, stream>>>(out, deg, n);
    gcn_degree<<<(E + 255) / 256, 256, 0, stream>>>(ei + E, deg, E);
    gcn_dinv<<<(n + 255) / 256, 256, 0, stream>>>(deg, dinv, n);
    gcn_gemm_wmma<<<(n + GEMM_ROWS - 1) / GEMM_ROWS, 256, 0, stream>>>(x, W, h, n);

    long long scatter_threads = (long long)E * OUT_DIM;
    int scatter_blocks = (int)((scatter_threads + 255) / 256);
    gcn_scatter<<<scatter_blocks, 256, 0, stream>>>(ei, h, dinv, out, E);

    gcn_finalize<<<(total_out + 255) / 256, 256, 0, stream>>>(h, dinv, b, out, n);
}